// AggregationRebuild_36223754175056
// MI455X (gfx1250) — compile-verified
//
#include <hip/hip_runtime.h>
#include <cstdint>

// Problem constants (from the reference)
#define TEMPERATURE   0.5f
#define N_ROWS        2048
#define K_NB          8
#define LD            4096          // L*D = 64*64
#define THREADS       256
#define CHUNK         (THREADS * 4) // floats per block chunk (float4 per thread)

// ---------------------------------------------------------------------------
// Fused kernel.
//   Per block (n = blockIdx.y, j-chunk = blockIdx.x):
//     1) gather sim[n, idx[n,k]] (uniform scalar loads), softmax over K=8
//        in registers (redundant across the 4 j-chunks of a row - 8 scalar
//        loads + ~30 VALU ops, far cheaper than a second kernel launch).
//     2) blockIdx.x==0, tid==0 writes the w[n,:] output tuple element.
//     3) 8 gathered async B128 copies global->LDS (CDNA5 ASYNCcnt engine,
//        GV addressing), one s_wait_asynccnt, then 8 ds_load_b128 + FMA,
//        one global_store_b128.
// Memory roofline: emb(32MB, L2-resident) read + out(32MB) written
//   => ~64MB HBM => ~2.8us floor at 23.3 TB/s. Everything else is noise.
// ---------------------------------------------------------------------------
__global__ void agg_rebuild_fused_kernel(const float* __restrict__ sim,
                                         const float* __restrict__ emb,
                                         const int*   __restrict__ index,
                                         float*       __restrict__ w_out,
                                         float*       __restrict__ out) {
    // 8 stage buffers of 4KB each = 32KB LDS (WGP has 320KB).
    __shared__ float lds_buf[K_NB * CHUNK];

    const int tid   = threadIdx.x;                 // 0..255
    const int n     = blockIdx.y;                  // row
    const int jbase = blockIdx.x * CHUNK + tid * 4;

    // ---- neighbor indices (uniform across block -> scalar loads) ----------
    int idxs[K_NB];
#pragma unroll
    for (int k = 0; k < K_NB; ++k)
        idxs[k] = index[n * K_NB + k];

    // ---- issue the gathered async copies as early as possible -------------
    // LDS byte address = low 32 bits of the generic LDS pointer
    // (ISA 10.2: LDS aperture flat addresses carry the offset in addr[31:0]).
#pragma unroll
    for (int k = 0; k < K_NB; ++k) {
        const uint64_t gaddr =
            (uint64_t)(uintptr_t)(emb + (size_t)idxs[k] * LD + jbase);
        const uint32_t laddr =
            (uint32_t)(uintptr_t)(&lds_buf[k * CHUNK + tid * 4]);
        asm volatile("global_load_async_to_lds_b128 %0, %1, off"
                     :: "v"(laddr), "v"(gaddr)
                     : "memory");
    }

    // ---- softmax weights (uniform, overlaps async copy latency) -----------
    const float* simrow = sim + (size_t)n * N_ROWS;
    float wk[K_NB];
    float m = -3.402823466e38f;
#pragma unroll
    for (int k = 0; k < K_NB; ++k) {
        wk[k] = simrow[idxs[k]] * (1.0f / TEMPERATURE);
        m = fmaxf(m, wk[k]);
    }
    float s = 0.0f;
#pragma unroll
    for (int k = 0; k < K_NB; ++k) {
        wk[k] = __expf(wk[k] - m);
        s += wk[k];
    }
    const float inv = 1.0f / s;
#pragma unroll
    for (int k = 0; k < K_NB; ++k)
        wk[k] *= inv;

    // ---- emit the w output tuple element once per row ---------------------
    if (blockIdx.x == 0 && tid == 0) {
#pragma unroll
        for (int k = 0; k < K_NB; ++k)
            w_out[n * K_NB + k] = wk[k];
    }

    // ---- wait for this wave's async copies (ASYNCcnt -> 0), accumulate ----
    asm volatile("s_wait_asynccnt 0" ::: "memory");

    float4 acc = make_float4(0.f, 0.f, 0.f, 0.f);
#pragma unroll
    for (int k = 0; k < K_NB; ++k) {
        const float4 f =
            *reinterpret_cast<const float4*>(&lds_buf[k * CHUNK + tid * 4]);
        acc.x = fmaf(wk[k], f.x, acc.x);
        acc.y = fmaf(wk[k], f.y, acc.y);
        acc.z = fmaf(wk[k], f.z, acc.z);
        acc.w = fmaf(wk[k], f.w, acc.w);
    }

    *reinterpret_cast<float4*>(out + (size_t)n * LD + jbase) = acc;
}

// ---------------------------------------------------------------------------
// Launch: d_out = [ w (N*K floats) | rebuilt emb (N*L*D floats) ]
// ---------------------------------------------------------------------------
extern "C" void kernel_launch(void* const* d_in, const int* in_sizes, int n_in,
                              void* d_out, int out_size, void* d_ws, size_t ws_size,
                              hipStream_t stream) {
    const float* sim   = (const float*)d_in[0];
    const float* emb   = (const float*)d_in[1];
    const int*   index = (const int*)d_in[2];

    float* w_out   = (float*)d_out;                      // [N, K]
    float* emb_out = (float*)d_out + N_ROWS * K_NB;      // [N, L*D]

    dim3 grid(LD / CHUNK, N_ROWS);                       // (4, 2048)
    agg_rebuild_fused_kernel<<<grid, THREADS, 0, stream>>>(sim, emb, index,
                                                           w_out, emb_out);
}